// MultiLinear_11115375362121
// MI455X (gfx1250) — compile-verified
//
#include <hip/hip_runtime.h>

#define B_SZ     4096
#define IN_DIMK  256
#define OUT_DIMN 256
#define NHEADS   16
#define ROWS_PB  32                 // rows (M) per block: 2 row-tiles of 16
#define XS_PITCH (IN_DIMK + 4)      // 260: row stride -> distinct LDS banks

typedef float v2f __attribute__((ext_vector_type(2)));
typedef float v8f __attribute__((ext_vector_type(8)));

// ---------------------------------------------------------------------------
// Pass 1: bucket samples by head id.  perm[] gets sample indices grouped by
// head; offsets[h]..offsets[h+1] is head h's slice.  Single block.
// ---------------------------------------------------------------------------
__global__ __launch_bounds__(256) void ml_bucket(const int* __restrict__ head_ids,
                                                 int* __restrict__ perm,
                                                 int* __restrict__ offsets) {
    __shared__ int cnt[NHEADS];
    __shared__ int base[NHEADS + 1];
    const int t = threadIdx.x;

    if (t < NHEADS) cnt[t] = 0;
    __syncthreads();

    for (int b = t; b < B_SZ; b += 256)
        atomicAdd(&cnt[head_ids[b]], 1);
    __syncthreads();

    if (t == 0) {
        int s = 0;
        for (int h = 0; h < NHEADS; ++h) { base[h] = s; s += cnt[h]; }
        base[NHEADS] = s;
    }
    __syncthreads();

    if (t <= NHEADS) offsets[t] = base[t];
    if (t < NHEADS)  cnt[t] = base[t];          // running cursor per head
    __syncthreads();

    for (int b = t; b < B_SZ; b += 256) {
        const int h   = head_ids[b];
        const int pos = atomicAdd(&cnt[h], 1);
        perm[pos] = b;
    }
}

// ---------------------------------------------------------------------------
// Pass 2: grouped GEMM.  Block = 32 gathered rows of one head x all 256 cols.
// 8 waves/block; each wave owns a 32x32 output patch = 4 independent 16x16
// WMMA accumulator chains (2 row-tiles x 2 col-tiles) sharing B fragments.
// K accumulated via v_wmma_f32_16x16x4_f32 (exact fp32).
// ---------------------------------------------------------------------------
__global__ __launch_bounds__(256) void ml_gemm(const float* __restrict__ x,
                                               const float* __restrict__ w,
                                               const float* __restrict__ bias,
                                               const int* __restrict__ perm,
                                               const int* __restrict__ offsets,
                                               float* __restrict__ out) {
    __shared__ float xs[ROWS_PB][XS_PITCH];     // 32 x 256 fp32 tile, padded
    __shared__ int   sperm[ROWS_PB];
    __shared__ int   srange[2];

    const int t    = threadIdx.x;
    const int head = blockIdx.y;

    if (t < 2) srange[t] = offsets[head + t];
    __syncthreads();
    const int start   = srange[0];
    const int end     = srange[1];
    const int rowbase = start + (int)blockIdx.x * ROWS_PB;
    if (rowbase >= end) return;                 // uniform for whole block
    const int m = min(ROWS_PB, end - rowbase);  // valid rows in this block

    if (t < ROWS_PB) sperm[t] = (t < m) ? perm[rowbase + t] : 0;
    __syncthreads();

    // ---- stage gathered X rows into LDS (8 x float4 per thread) ----
    {
        const int row = t & 31;                 // 32 rows
        const int c0  = (t >> 5) * 32;          // 8 threads/row, 32 cols each
        if (row < m) {
            const float* src = x + (size_t)sperm[row] * IN_DIMK + c0;
            #pragma unroll
            for (int j = 0; j < 8; ++j) {
                float4 v = *(const float4*)(src + j * 4);
                *(float4*)&xs[row][c0 + j * 4] = v;
            }
        } else {
            const float4 z = make_float4(0.f, 0.f, 0.f, 0.f);
            #pragma unroll
            for (int j = 0; j < 8; ++j)
                *(float4*)&xs[row][c0 + j * 4] = z;
        }
    }
    __syncthreads();

    const int lane = t & 31;
    const int wave = t >> 5;
    const int half = lane >> 4;                 // 0: {K,K+1}  1: {K+2,K+3}
    const int ln   = lane & 15;                 // M for A-frag, N for B-frag

    const int col0 = (wave * 2 + 0) * 16 + ln;  // wave's two 16-col tiles
    const int col1 = (wave * 2 + 1) * 16 + ln;  // = col0 + 16

    const float* wh    = w + (size_t)head * IN_DIMK * OUT_DIMN;
    const float* wcol0 = wh + col0;             // walk K rows, stride OUT_DIMN

    v8f acc00 = {0.f,0.f,0.f,0.f,0.f,0.f,0.f,0.f};  // [row-tile][col-tile]
    v8f acc01 = {0.f,0.f,0.f,0.f,0.f,0.f,0.f,0.f};
    v8f acc10 = {0.f,0.f,0.f,0.f,0.f,0.f,0.f,0.f};
    v8f acc11 = {0.f,0.f,0.f,0.f,0.f,0.f,0.f,0.f};

    #pragma unroll 4
    for (int ks = 0; ks < IN_DIMK / 4; ++ks) {
        const int kk = 4 * ks + 2 * half;
        // A frags: 16x4 fp32, lane=M; VGPR0/1 = {K,K+1} (lanes 0-15) or
        // {K+2,K+3} (lanes 16-31).  One ds_load_b64 per row-tile.
        const float2 a0_ = *(const float2*)&xs[ln][kk];
        const float2 a1_ = *(const float2*)&xs[ln + 16][kk];
        v2f a0; a0.x = a0_.x; a0.y = a0_.y;
        v2f a1; a1.x = a1_.x; a1.y = a1_.y;
        // B frags: 4x16 fp32, lane=N; same K split; shared by both row-tiles.
        v2f b0, b1;
        b0.x = wcol0[(size_t)kk * OUT_DIMN];
        b0.y = wcol0[(size_t)(kk + 1) * OUT_DIMN];
        b1.x = wcol0[(size_t)kk * OUT_DIMN + 16];
        b1.y = wcol0[(size_t)(kk + 1) * OUT_DIMN + 16];
        // 4 independent accumulator chains
        acc00 = __builtin_amdgcn_wmma_f32_16x16x4_f32(false, a0, false, b0,
                                                      (short)0, acc00, false, false);
        acc10 = __builtin_amdgcn_wmma_f32_16x16x4_f32(false, a1, false, b0,
                                                      (short)0, acc10, false, false);
        acc01 = __builtin_amdgcn_wmma_f32_16x16x4_f32(false, a0, false, b1,
                                                      (short)0, acc01, false, false);
        acc11 = __builtin_amdgcn_wmma_f32_16x16x4_f32(false, a1, false, b1,
                                                      (short)0, acc11, false, false);
    }

    const float bv0 = bias[head * OUT_DIMN + col0];
    const float bv1 = bias[head * OUT_DIMN + col1];

    // C/D layout: VGPR r -> M=r (lanes 0-15) / M=r+8 (lanes 16-31), N=ln.
    // One sperm read per row, reused by both column tiles.
    #pragma unroll
    for (int rt = 0; rt < 2; ++rt) {
        #pragma unroll
        for (int r = 0; r < 8; ++r) {
            const int rm = rt * 16 + r + 8 * half;
            if (rm < m) {
                float* o = out + (size_t)sperm[rm] * OUT_DIMN;
                const float c0v = (rt == 0) ? acc00[r] : acc10[r];
                const float c1v = (rt == 0) ? acc01[r] : acc11[r];
                o[col0] = c0v + bv0;
                o[col1] = c1v + bv1;
            }
        }
    }
}

// ---------------------------------------------------------------------------
// Launch: inputs order = {inputs, head_ids, weights, biases}
// ---------------------------------------------------------------------------
extern "C" void kernel_launch(void* const* d_in, const int* in_sizes, int n_in,
                              void* d_out, int out_size, void* d_ws, size_t ws_size,
                              hipStream_t stream) {
    const float* inputs   = (const float*)d_in[0];
    const int*   head_ids = (const int*)  d_in[1];
    const float* weights  = (const float*)d_in[2];
    const float* biases   = (const float*)d_in[3];
    float*       out      = (float*)d_out;

    int* perm    = (int*)d_ws;            // B_SZ ints
    int* offsets = perm + B_SZ;           // NHEADS+1 ints

    hipLaunchKernelGGL(ml_bucket, dim3(1), dim3(256), 0, stream,
                       head_ids, perm, offsets);

    dim3 grid(B_SZ / ROWS_PB, NHEADS);    // worst case: one head owns all rows
    hipLaunchKernelGGL(ml_gemm, grid, dim3(256), 0, stream,
                       inputs, weights, biases, perm, offsets, out);
}